// FGNet_37752762531919
// MI455X (gfx1250) — compile-verified
//
#include <hip/hip_runtime.h>

#define MAX_ATOMS  13
#define NUM_PARAMS 169
#define L_DIM      64
#define R_DIM      128
#define ORDER      3
#define TILE_M     16

typedef float v2f __attribute__((ext_vector_type(2)));
typedef float v8f __attribute__((ext_vector_type(8)));

// ---------------- pass 0: zero output + counters ----------------
__global__ void fg_zero(float* __restrict__ out, int outN, int* __restrict__ counts) {
    int i = blockIdx.x * blockDim.x + threadIdx.x;
    if (i < outN) out[i] = 0.0f;
    if (i < NUM_PARAMS) counts[i] = 0;
}

// ---------------- pass 1: per-id histogram + edge ids ----------------
__global__ void fg_hist(const int* __restrict__ x, const int* __restrict__ fact, int E,
                        int* __restrict__ counts, int* __restrict__ edge_id) {
    int e = blockIdx.x * blockDim.x + threadIdx.x;
    if (e >= E) return;
    int f0 = fact[e * ORDER + 0];
    int id = x[f0 * 3 + 1] * MAX_ATOMS + x[f0 * 3 + 2];
    edge_id[e] = id;
    atomicAdd(&counts[id], 1);
}

// ---------------- pass 2: serial scan over 169 ids + tile worklist ----------------
__global__ void fg_scan(const int* __restrict__ counts, int* __restrict__ offsets,
                        int* __restrict__ cursor, int* __restrict__ tile_id,
                        int* __restrict__ tile_start, int* __restrict__ tile_cnt,
                        int* __restrict__ numTiles) {
    if (threadIdx.x != 0 || blockIdx.x != 0) return;
    int off = 0, nt = 0;
    for (int i = 0; i < NUM_PARAMS; ++i) {
        offsets[i] = off;
        cursor[i]  = off;
        int c = counts[i];
        int t = (c + TILE_M - 1) / TILE_M;
        for (int k = 0; k < t; ++k) {
            tile_id[nt]    = i;
            tile_start[nt] = off + k * TILE_M;
            int rem = c - k * TILE_M;
            tile_cnt[nt]   = rem < TILE_M ? rem : TILE_M;
            ++nt;
        }
        off += c;
    }
    offsets[NUM_PARAMS] = off;
    numTiles[0] = nt;
}

// ---------------- pass 3: scatter edges into id buckets ----------------
__global__ void fg_scatter(const int* __restrict__ edge_id, int E,
                           int* __restrict__ cursor, int* __restrict__ bucket) {
    int e = blockIdx.x * blockDim.x + threadIdx.x;
    if (e >= E) return;
    int pos = atomicAdd(&cursor[edge_id[e]], 1);
    bucket[pos] = e;
}

// ---------------- pass 4: fused WMMA compute per 16-edge tile ----------------
__global__ __launch_bounds__(128) void fg_main(
    const float* __restrict__ nodes,   const int* __restrict__ fact,
    const float* __restrict__ params,  const float* __restrict__ bias_p,
    const float* __restrict__ ho_params, const float* __restrict__ ho_bias,
    const int* __restrict__ tile_id,   const int* __restrict__ tile_start,
    const int* __restrict__ tile_cnt,  const int* __restrict__ numTiles,
    const int* __restrict__ bucket,    float* __restrict__ out)
{
    __shared__ float sNodes[ORDER][TILE_M][L_DIM + 1];   // +1 pad: avoid 16-way bank conflict
    __shared__ float sT[ORDER][TILE_M][R_DIM + 1];       // staged transforms (post-ReLU)
    __shared__ int   sFact[TILE_M][ORDER];

    const int bt = blockIdx.x;
    if (bt >= numTiles[0]) return;                        // wave-uniform exit (EXEC stays full)
    const int id    = tile_id[bt];
    const int start = tile_start[bt];
    const int cnt   = tile_cnt[bt];
    const int tid   = threadIdx.x;
    const int wave  = tid >> 5;
    const int lane  = tid & 31;

    // fact rows for this tile (dummy 0 for padded rows)
    if (tid < TILE_M * ORDER) {
        int m = tid / ORDER, o = tid % ORDER;
        int v = 0;
        if (m < cnt) {
            int e = bucket[start + m];
            v = fact[e * ORDER + o];
        }
        sFact[m][o] = v;
    }
    __syncthreads();

    // gather node rows into LDS (zero padded rows so WMMA A-rows are clean)
    for (int q = tid; q < ORDER * TILE_M * L_DIM; q += 128) {
        int o   = q / (TILE_M * L_DIM);
        int rem = q - o * (TILE_M * L_DIM);
        int m   = rem / L_DIM;
        int l   = rem - m * L_DIM;
        float v = 0.0f;
        if (m < cnt) v = nodes[(size_t)sFact[m][o] * L_DIM + l];
        sNodes[o][m][l] = v;
    }
    __syncthreads();

    const int mrow  = lane & 15;
    const int khalf = (lane >> 4) << 1;   // lanes 0-15 -> K+0/K+1, lanes 16-31 -> K+2/K+3
    const int mbase = (lane >> 4) << 3;   // C/D rows: lo lanes M=0..7, hi lanes M=8..15

    // ---- Phase A: T[o] = relu(nodes_tile @ W + b), 24 (o, n-tile) units over 4 waves ----
    const float* __restrict__ W = params + (size_t)id * L_DIM * R_DIM;  // (64,128) row-major
    for (int u = wave; u < ORDER * (R_DIM / 16); u += 4) {
        int o = u >> 3;
        int n = (u & 7) * 16 + (lane & 15);
        v8f acc = {};
        #pragma unroll
        for (int k0 = 0; k0 < L_DIM; k0 += 4) {
            int ka = k0 + khalf;
            v2f a, b;
            a.x = sNodes[o][mrow][ka];
            a.y = sNodes[o][mrow][ka + 1];
            b.x = W[(size_t)ka * R_DIM + n];
            b.y = W[(size_t)(ka + 1) * R_DIM + n];
            acc = __builtin_amdgcn_wmma_f32_16x16x4_f32(false, a, false, b,
                                                        (short)0, acc, false, false);
        }
        float bias = bias_p[(size_t)id * R_DIM + n];
        #pragma unroll
        for (int v = 0; v < 8; ++v) {
            float t = acc[v] + bias;
            sT[o][mbase + v][n] = t > 0.0f ? t : 0.0f;
        }
    }
    __syncthreads();

    // ---- Phase B: msg_i = (T_j1 * T_j2) @ W2_i + b2, scatter-add; 12 (i, l-tile) units ----
    for (int u = wave; u < ORDER * (L_DIM / 16); u += 4) {
        int i  = u >> 2;
        int j1 = (i == 0) ? 1 : 0;
        int j2 = (i == 2) ? 1 : 2;
        int lcol = (u & 3) * 16 + (lane & 15);
        const float* __restrict__ W2 =
            ho_params + ((size_t)i * NUM_PARAMS + id) * (R_DIM * L_DIM); // (128,64) row-major
        v8f acc = {};
        #pragma unroll
        for (int k0 = 0; k0 < R_DIM; k0 += 4) {
            int ka = k0 + khalf;
            v2f a, b;
            a.x = sT[j1][mrow][ka]     * sT[j2][mrow][ka];
            a.y = sT[j1][mrow][ka + 1] * sT[j2][mrow][ka + 1];
            b.x = W2[(size_t)ka * L_DIM + lcol];
            b.y = W2[(size_t)(ka + 1) * L_DIM + lcol];
            acc = __builtin_amdgcn_wmma_f32_16x16x4_f32(false, a, false, b,
                                                        (short)0, acc, false, false);
        }
        float hb = ho_bias[((size_t)i * NUM_PARAMS + id) * L_DIM + lcol];
        #pragma unroll
        for (int v = 0; v < 8; ++v) {
            int m = mbase + v;
            if (m < cnt) {
                int dest = sFact[m][i];
                atomicAdd(&out[(size_t)dest * L_DIM + lcol], acc[v] + hb);
            }
        }
    }
}

// ---------------- launcher ----------------
extern "C" void kernel_launch(void* const* d_in, const int* in_sizes, int n_in,
                              void* d_out, int out_size, void* d_ws, size_t ws_size,
                              hipStream_t stream) {
    const int*   x         = (const int*)  d_in[0];
    const float* nodes     = (const float*)d_in[1];
    const int*   fact      = (const int*)  d_in[2];
    // d_in[3] = fact_dim scalar (== ORDER), unused
    const float* params    = (const float*)d_in[4];
    const float* bias_p    = (const float*)d_in[5];
    const float* ho_params = (const float*)d_in[6];
    const float* ho_bias   = (const float*)d_in[7];
    float*       out       = (float*)d_out;

    const int E    = in_sizes[2] / ORDER;
    const int outN = out_size;               // N * L_DIM

    // workspace carve-up (ints, ~360 KB total)
    char* ws = (char*)d_ws;
    int* counts     = (int*)ws; ws += NUM_PARAMS * sizeof(int);
    int* offsets    = (int*)ws; ws += (NUM_PARAMS + 1) * sizeof(int);
    int* cursor     = (int*)ws; ws += NUM_PARAMS * sizeof(int);
    int* numTiles   = (int*)ws; ws += sizeof(int);
    const int maxTiles = E / TILE_M + NUM_PARAMS + 2;
    int* tile_id    = (int*)ws; ws += maxTiles * sizeof(int);
    int* tile_start = (int*)ws; ws += maxTiles * sizeof(int);
    int* tile_cnt   = (int*)ws; ws += maxTiles * sizeof(int);
    int* edge_id    = (int*)ws; ws += E * sizeof(int);
    int* bucket     = (int*)ws; ws += E * sizeof(int);

    fg_zero<<<(outN + 255) / 256, 256, 0, stream>>>(out, outN, counts);
    fg_hist<<<(E + 255) / 256, 256, 0, stream>>>(x, fact, E, counts, edge_id);
    fg_scan<<<1, 32, 0, stream>>>(counts, offsets, cursor,
                                  tile_id, tile_start, tile_cnt, numTiles);
    fg_scatter<<<(E + 255) / 256, 256, 0, stream>>>(edge_id, E, cursor, bucket);
    fg_main<<<maxTiles, 128, 0, stream>>>(nodes, fact, params, bias_p, ho_params, ho_bias,
                                          tile_id, tile_start, tile_cnt, numTiles,
                                          bucket, out);
}